// MaxofErosions2D_489626272308
// MI455X (gfx1250) — compile-verified
//
#include <hip/hip_runtime.h>

// Problem constants (match reference)
#define Bn   32
#define Hn   256
#define Wn   256
#define Cn   3
#define KHn  5
#define KWn  5
#define Fn   8

#define TILE 16
#define HALO 2
#define TW   (TILE + 2*HALO)   // 20
#define SROW (TW * Cn)         // 60 floats per LDS tile row

// gfx1250 async copy: global memory -> LDS, tracked by ASYNCcnt.
// VDST vgpr carries the LDS byte address; GV mode (64-bit vaddr, saddr=off).
__device__ __forceinline__ void async_ld_f32_to_lds(unsigned lds_byte_off,
                                                    const float* gaddr) {
  asm volatile("global_load_async_to_lds_b32 %0, %1, off"
               :: "v"(lds_byte_off), "v"(gaddr)
               : "memory");
}

__device__ __forceinline__ void wait_asynccnt0() {
  asm volatile("s_wait_asynccnt 0" ::: "memory");
}

__global__ __launch_bounds__(256)
void MaxofErosions2D_489626272308_kernel(const float* __restrict__ x,
                                         const float* __restrict__ kern,
                                         float* __restrict__ out) {
  // 20x20x3 input tile (+inf halo) and 5x5x3x8 kernel staged in LDS.
  __shared__ __align__(16) float sx[TW * TW * Cn];          // 1200 f32 = 4.8 KB
  __shared__ __align__(16) float sk[KHn * KWn * Cn * Fn];   //  600 f32 = 2.4 KB

  const int tid = threadIdx.x;
  const int bx = blockIdx.x, by = blockIdx.y, b = blockIdx.z;
  const int h0 = by * TILE, w0 = bx * TILE;

  // ---- Stage kernel weights (always in-bounds) via async copies ----
  for (int i = tid; i < KHn * KWn * Cn * Fn; i += 256) {
    unsigned off = (unsigned)(uintptr_t)(&sk[i]);
    async_ld_f32_to_lds(off, kern + i);
  }

  // ---- Stage input tile with +inf padding outside the image ----
  for (int i = tid; i < TW * TW * Cn; i += 256) {
    const int row = i / SROW;
    const int rem = i - row * SROW;
    const int col = rem / Cn;
    const int c   = rem - col * Cn;
    const int gh  = h0 + row - HALO;
    const int gw  = w0 + col - HALO;
    unsigned off = (unsigned)(uintptr_t)(&sx[i]);
    if ((unsigned)gh < (unsigned)Hn && (unsigned)gw < (unsigned)Wn) {
      const float* g = x + ((((size_t)b * Hn + gh) * Wn + gw) * Cn + c);
      async_ld_f32_to_lds(off, g);        // EXEC-predicated per-lane async copy
    } else {
      sx[i] = __builtin_inff();           // ds_store of +inf halo
    }
  }

  wait_asynccnt0();     // this wave's async copies landed in LDS
  __syncthreads();      // all waves' data visible

  const int ly = tid >> 4;     // 0..15
  const int lx = tid & 15;     // 0..15

  float acc[Fn];
#pragma unroll
  for (int f = 0; f < Fn; ++f) acc[f] = -__builtin_inff();

  // out = max over channels of ( min over taps of (x - kflip) )
  for (int c = 0; c < Cn; ++c) {
    float ero[Fn];
#pragma unroll
    for (int f = 0; f < Fn; ++f) ero[f] = __builtin_inff();

    for (int dy = 0; dy < KHn; ++dy) {
#pragma unroll
      for (int dx = 0; dx < KWn; ++dx) {
        const float v = sx[((ly + dy) * TW + (lx + dx)) * Cn + c];
        // kernel flipped: kflip[dy][dx] = kern[KH-1-dy][KW-1-dx]
        const float* kp =
            &sk[(((KHn - 1 - dy) * KWn + (KWn - 1 - dx)) * Cn + c) * Fn];
#pragma unroll
        for (int f = 0; f < Fn; ++f)
          ero[f] = fminf(ero[f], v - kp[f]);   // sub + v_min (min3-fusible)
      }
    }
#pragma unroll
    for (int f = 0; f < Fn; ++f) acc[f] = fmaxf(acc[f], ero[f]);
  }

  // ---- Store 8 filter outputs: two 128-bit global stores ----
  const int h = h0 + ly, w = w0 + lx;
  float4* o = (float4*)(out + (((size_t)b * Hn + h) * Wn + w) * Fn);
  o[0] = make_float4(acc[0], acc[1], acc[2], acc[3]);
  o[1] = make_float4(acc[4], acc[5], acc[6], acc[7]);
}

extern "C" void kernel_launch(void* const* d_in, const int* in_sizes, int n_in,
                              void* d_out, int out_size, void* d_ws, size_t ws_size,
                              hipStream_t stream) {
  const float* x    = (const float*)d_in[0];   // (32,256,256,3) f32
  const float* kern = (const float*)d_in[1];   // (5,5,3,8) f32
  float* out        = (float*)d_out;           // (32,256,256,8) f32

  dim3 grid(Wn / TILE, Hn / TILE, Bn);         // (16,16,32)
  dim3 block(256);                              // 8 wave32 per WGP slot
  MaxofErosions2D_489626272308_kernel<<<grid, block, 0, stream>>>(x, kern, out);
}